// GTN_Model_73761768341735
// MI455X (gfx1250) — compile-verified
//
#include <hip/hip_runtime.h>
#include <hip/hip_bf16.h>

typedef float v2f __attribute__((ext_vector_type(2)));
typedef float v8f __attribute__((ext_vector_type(8)));

#define BM 128
#define BN 128
#define BK 32

// ---------------------------------------------------------------------------
// Softmax of the three [2,5] weight matrices. Writes alphas to ws and the 30
// softmax outputs (Ws[0],Ws[1],Ws[2]) to the tail of d_out.
// ---------------------------------------------------------------------------
__global__ void k_softmax(const float* __restrict__ w0a,
                          const float* __restrict__ w0b,
                          const float* __restrict__ w1,
                          float* __restrict__ alpha,
                          float* __restrict__ out_tail) {
  if (threadIdx.x == 0) {
    const float* ws3[3] = {w0a, w0b, w1};
    for (int t = 0; t < 3; ++t) {
      for (int c = 0; c < 2; ++c) {
        const float* p = ws3[t] + c * 5;
        float mx = p[0];
        for (int e = 1; e < 5; ++e) mx = fmaxf(mx, p[e]);
        float ex[5], s = 0.f;
        for (int e = 0; e < 5; ++e) { ex[e] = expf(p[e] - mx); s += ex[e]; }
        for (int e = 0; e < 5; ++e) {
          float a = ex[e] / s;
          alpha[t * 10 + c * 5 + e] = a;
          out_tail[t * 10 + c * 5 + e] = a;
        }
      }
    }
  }
}

// ---------------------------------------------------------------------------
// GTConv fused: one pass over A[N,N,5] -> a0[c], b0[c], a1[c] planes.
// alpha layout: [t(3)][c(2)][e(5)]
// ---------------------------------------------------------------------------
__global__ __launch_bounds__(256)
void k_gtconv(const float* __restrict__ A, const float* __restrict__ alpha,
              float* __restrict__ a0, float* __restrict__ b0,
              float* __restrict__ a1, long total) {
  long idx = (long)blockIdx.x * 256 + threadIdx.x;
  if (idx >= total) return;
  const float* p = A + idx * 5;
  float v0 = p[0], v1 = p[1], v2 = p[2], v3 = p[3], v4 = p[4];
#pragma unroll
  for (int c = 0; c < 2; ++c) {
    const float* aa = alpha + c * 5;
    const float* ab = alpha + 10 + c * 5;
    const float* ac = alpha + 20 + c * 5;
    a0[c * total + idx] = aa[0]*v0 + aa[1]*v1 + aa[2]*v2 + aa[3]*v3 + aa[4]*v4;
    b0[c * total + idx] = ab[0]*v0 + ab[1]*v1 + ab[2]*v2 + ab[3]*v3 + ab[4]*v4;
    a1[c * total + idx] = ac[0]*v0 + ac[1]*v1 + ac[2]*v2 + ac[3]*v3 + ac[4]*v4;
  }
}

// ---------------------------------------------------------------------------
// WMMA fp32 GEMM: C = op(A) * B (+bias)(+relu), row-major, batched via grid.z.
//   !TRANSA: A is [M,K] (lda=K).  TRANSA: Aphys is [K,M] (lda=M), A[m,k]=Aphys[k,m].
//   B is [K,N] (ldb=N). C is [M,N]. M%128==0, N%128==0, K%32==0 required.
// Block: 256 threads = 8 waves; macro tile 128x128; wave -> 64x32 (4x2 WMMA tiles).
// ---------------------------------------------------------------------------
template <bool TRANSA, bool BIAS, bool RELU>
__global__ __launch_bounds__(256)
void gemm_wmma(const float* __restrict__ A, const float* __restrict__ B,
               const float* __restrict__ bias, float* __restrict__ C,
               int M, int Nn, int K, long sA, long sB, long sC) {
  __shared__ float As[BM][BK + 4];
  __shared__ float Bs[BK][BN + 4];

  const int tid = threadIdx.x;
  const int lane = tid & 31;
  const int w = tid >> 5;
  const int wm = w >> 2;             // 0..1
  const int wn = w & 3;              // 0..3
  const int laneHalf = lane >> 4;    // 0/1
  const int lane15 = lane & 15;
  const int blockRow = blockIdx.y * BM;
  const int blockCol = blockIdx.x * BN;

  A += (long)blockIdx.z * sA;
  B += (long)blockIdx.z * sB;
  C += (long)blockIdx.z * sC;

  const v8f vzero = {0.f, 0.f, 0.f, 0.f, 0.f, 0.f, 0.f, 0.f};
  v8f acc[4][2];
#pragma unroll
  for (int i = 0; i < 4; ++i)
#pragma unroll
    for (int j = 0; j < 2; ++j) acc[i][j] = vzero;

  for (int kt = 0; kt < K; kt += BK) {
    if (!TRANSA) {
      // 128 rows x 32 cols = 1024 float4, 4 per thread
#pragma unroll
      for (int s = 0; s < 4; ++s) {
        int idx4 = tid + s * 256;
        int row = idx4 >> 3;
        int c4 = (idx4 & 7) << 2;
        float4 v = *(const float4*)(A + (long)(blockRow + row) * K + kt + c4);
        *(float4*)(&As[row][c4]) = v;
      }
    } else {
      // Aphys[k, m]: 32 k-rows x 128 m = 1024 float4 (over m), transpose into As
#pragma unroll
      for (int s = 0; s < 4; ++s) {
        int idx4 = tid + s * 256;
        int kk = idx4 >> 5;
        int m4 = (idx4 & 31) << 2;
        float4 v = *(const float4*)(A + (long)(kt + kk) * M + blockRow + m4);
        As[m4 + 0][kk] = v.x;
        As[m4 + 1][kk] = v.y;
        As[m4 + 2][kk] = v.z;
        As[m4 + 3][kk] = v.w;
      }
    }
    // B tile: 32 k-rows x 128 n
#pragma unroll
    for (int s = 0; s < 4; ++s) {
      int idx4 = tid + s * 256;
      int kk = idx4 >> 5;
      int n4 = (idx4 & 31) << 2;
      float4 v = *(const float4*)(B + (long)(kt + kk) * Nn + blockCol + n4);
      *(float4*)(&Bs[kk][n4]) = v;
    }
    __syncthreads();

#pragma unroll
    for (int k8 = 0; k8 < 8; ++k8) {
      // per-lane K offsets for 16x16x4 fp32 WMMA: lanes 0-15 -> {0,1}, 16-31 -> {2,3}
      const int kA = (k8 << 2) + (laneHalf << 1);
      v2f afr[4], bfr[2];
#pragma unroll
      for (int sm = 0; sm < 4; ++sm) {
        int r = wm * 64 + sm * 16 + lane15;
        afr[sm].x = As[r][kA];
        afr[sm].y = As[r][kA + 1];
      }
#pragma unroll
      for (int sn = 0; sn < 2; ++sn) {
        int cc = wn * 32 + sn * 16 + lane15;
        bfr[sn].x = Bs[kA][cc];
        bfr[sn].y = Bs[kA + 1][cc];
      }
#pragma unroll
      for (int sm = 0; sm < 4; ++sm)
#pragma unroll
        for (int sn = 0; sn < 2; ++sn)
          acc[sm][sn] = __builtin_amdgcn_wmma_f32_16x16x4_f32(
              false, afr[sm], false, bfr[sn], (short)0, acc[sm][sn], false, false);
    }
    __syncthreads();
  }

  // Epilogue: C/D layout — VGPR v holds row (v + 8*laneHalf), col = lane15
#pragma unroll
  for (int sm = 0; sm < 4; ++sm)
#pragma unroll
    for (int sn = 0; sn < 2; ++sn) {
      int col = blockCol + wn * 32 + sn * 16 + lane15;
      float bv = BIAS ? bias[col] : 0.f;
#pragma unroll
      for (int v = 0; v < 8; ++v) {
        int row = blockRow + wm * 64 + sm * 16 + (laneHalf << 3) + v;
        float val = acc[sm][sn][v] + bv;
        if (RELU) val = fmaxf(val, 0.f);
        C[(long)row * Nn + col] = val;
      }
    }
}

// ---------------------------------------------------------------------------
// Column sums excluding diagonal (+1 if ADD1) -> dinv, diag per (c, m)
// ---------------------------------------------------------------------------
template <bool ADD1>
__global__ __launch_bounds__(256)
void k_colsum(const float* __restrict__ H, float* __restrict__ dinv,
              float* __restrict__ diag, int n) {
  int m = blockIdx.x * 256 + threadIdx.x;
  int c = blockIdx.y;
  const float* Hc = H + (long)c * n * n;
  float s = 0.f;
  for (int r = 0; r < n; ++r) s += Hc[(long)r * n + m];
  float d = Hc[(long)m * n + m];
  float deg = s - d + (ADD1 ? 1.f : 0.f);
  dinv[c * n + m] = (deg == 0.f) ? 0.f : 1.f / deg;
  diag[c * n + m] = d;
}

// In-place norm(add=false): zero diag, scale column m by dinv[m]
__global__ __launch_bounds__(256)
void k_scale0(float* __restrict__ H, const float* __restrict__ dinv, int n) {
  int idx = blockIdx.x * 256 + threadIdx.x;  // < n*n (16.7M fits in int)
  int c = blockIdx.y;
  int rr = idx / n;
  int mm = idx - rr * n;
  float* Hc = H + (long)c * n * n;
  float v = Hc[idx];
  Hc[idx] = (rr == mm) ? 0.f : v * dinv[c * n + mm];
}

// Fold norm(add=true) into the T = H1^T @ XW result, relu, channel-interleave.
// X_[nn, c*F+f] = relu( dinv1[c,nn] * (T[c,nn,f] + (1 - diag1[c,nn]) * XW[nn,f]) )
__global__ __launch_bounds__(256)
void k_fixup(const float* __restrict__ T, const float* __restrict__ XW,
             const float* __restrict__ dinv1, const float* __restrict__ diag1,
             float* __restrict__ Xcat, int n, int F, int Cc) {
  int idx = blockIdx.x * 256 + threadIdx.x;  // total Cc*n*F
  int f = idx % F;
  int rem = idx / F;
  int nn = rem % n;
  int c = rem / n;
  float t = T[((long)c * n + nn) * F + f];
  float xw = XW[(long)nn * F + f];
  float val = dinv1[c * n + nn] * (t + (1.f - diag1[c * n + nn]) * xw);
  Xcat[(long)nn * (Cc * F) + c * F + f] = fmaxf(val, 0.f);
}

// y[t, j] = h1[target[t]] . lin2_w[:, j] + lin2_b[j]
__global__ __launch_bounds__(256)
void k_final(const float* __restrict__ h1, const int* __restrict__ tgt,
             const float* __restrict__ w2, const float* __restrict__ b2,
             float* __restrict__ y, int ntgt, int F, int ncls) {
  int idx = blockIdx.x * 256 + threadIdx.x;
  if (idx >= ntgt * ncls) return;
  int t = idx / ncls, j = idx - (idx / ncls) * ncls;
  int row = tgt[t];
  float s = b2[j];
  for (int f = 0; f < F; ++f) s += h1[(long)row * F + f] * w2[f * ncls + j];
  y[idx] = s;
}

// ---------------------------------------------------------------------------
extern "C" void kernel_launch(void* const* d_in, const int* in_sizes, int n_in,
                              void* d_out, int out_size, void* d_ws, size_t ws_size,
                              hipStream_t stream) {
  const float* A      = (const float*)d_in[0];
  const float* X      = (const float*)d_in[1];
  const int*   tgt    = (const int*)d_in[2];
  const float* w0a    = (const float*)d_in[3];
  const float* w0b    = (const float*)d_in[4];
  const float* w1     = (const float*)d_in[5];
  const float* gcn_w  = (const float*)d_in[6];
  const float* lin1_w = (const float*)d_in[7];
  const float* lin1_b = (const float*)d_in[8];
  const float* lin2_w = (const float*)d_in[9];
  const float* lin2_b = (const float*)d_in[10];
  float* out = (float*)d_out;

  const int n = 4096, Cc = 2, F = 128, WIN = 256, NCLS = 8, NTGT = 2000;
  const long nn = (long)n * n;

  float* ws    = (float*)d_ws;
  float* alpha = ws;               // 32
  float* a0    = ws + 32;          // 2*nn   (after H0 gemm: reused for XW/T/Xcat/h1)
  float* b0    = a0 + 2 * nn;      // 2*nn   (after H0 gemm: holds H1)
  float* a1    = b0 + 2 * nn;      // 2*nn
  float* h0    = a1 + 2 * nn;      // 2*nn   (scaled in place -> Hn)
  float* dinv0 = h0 + 2 * nn;      // 2*n
  float* diag0 = dinv0 + 2 * n;
  float* dinv1 = diag0 + 2 * n;
  float* diag1 = dinv1 + 2 * n;
  // overlays inside a0 region (free after H0 gemm); total 3.15M floats << 2*nn
  float* XW   = a0;                          // n*F
  float* T    = XW + (long)n * F;            // Cc*n*F
  float* Xcat = T + (long)Cc * n * F;        // n*Cc*F
  float* h1   = Xcat + (long)n * Cc * F;     // n*F

  // 1) softmax alphas (+ write the three Ws outputs to d_out tail)
  k_softmax<<<1, 32, 0, stream>>>(w0a, w0b, w1, alpha, out + NTGT * NCLS);
  // 2) fused GTConv: A -> a0, b0, a1 planes
  k_gtconv<<<(int)(nn / 256), 256, 0, stream>>>(A, alpha, a0, b0, a1, nn);
  // 3) H0 = a0 @ b0 per channel
  dim3 gSq(n / BN, n / BM, Cc);
  gemm_wmma<false, false, false><<<gSq, 256, 0, stream>>>(a0, b0, nullptr, h0,
                                                          n, n, n, nn, nn, nn);
  // 4) norm(add=false) prep + in-place scale -> Hn
  k_colsum<false><<<dim3(n / 256, Cc), 256, 0, stream>>>(h0, dinv0, diag0, n);
  k_scale0<<<dim3((int)(nn / 256), Cc), 256, 0, stream>>>(h0, dinv0, n);
  // 5) XW = X @ gcn_w  (a0 region free now)
  gemm_wmma<false, false, false><<<dim3(F / BN, n / BM, 1), 256, 0, stream>>>(
      X, gcn_w, nullptr, XW, n, F, WIN, 0, 0, 0);
  //    H1 = Hn @ a1 -> b0
  gemm_wmma<false, false, false><<<gSq, 256, 0, stream>>>(h0, a1, nullptr, b0,
                                                          n, n, n, nn, nn, nn);
  // 6) norm(add=true) prep on H1, then T = H1^T @ XW (batched over channels)
  k_colsum<true><<<dim3(n / 256, Cc), 256, 0, stream>>>(b0, dinv1, diag1, n);
  gemm_wmma<true, false, false><<<dim3(F / BN, n / BM, Cc), 256, 0, stream>>>(
      b0, XW, nullptr, T, n, F, n, nn, 0, (long)n * F);
  k_fixup<<<(Cc * n * F) / 256, 256, 0, stream>>>(T, XW, dinv1, diag1, Xcat, n, F, Cc);
  // 7) h1 = relu(Xcat @ lin1_w + lin1_b)
  gemm_wmma<false, true, true><<<dim3(F / BN, n / BM, 1), 256, 0, stream>>>(
      Xcat, lin1_w, lin1_b, h1, n, F, Cc * F, 0, 0, 0);
  // 8) y = h1[target] @ lin2_w + lin2_b
  k_final<<<(NTGT * NCLS + 255) / 256, 256, 0, stream>>>(h1, tgt, lin2_w,
                                                         lin2_b, out, NTGT, F, NCLS);
}